// ImprovedGATModel_20864951124558
// MI455X (gfx1250) — compile-verified
//
#include <hip/hip_runtime.h>
#include <hip/hip_bf16.h>
#include <math.h>

typedef float v2f __attribute__((ext_vector_type(2)));
typedef float v8f __attribute__((ext_vector_type(8)));

#define HID 64
#define TPB 256

__device__ __forceinline__ unsigned f2ord(float f) {
  unsigned u = __float_as_uint(f);
  return (u & 0x80000000u) ? ~u : (u | 0x80000000u);
}
__device__ __forceinline__ float ord2f(unsigned u) {
  return (u & 0x80000000u) ? __uint_as_float(u ^ 0x80000000u)
                           : __uint_as_float(~u);
}
__device__ __forceinline__ float elu1(float x) { return x > 0.f ? x : expm1f(x); }

// ---------------------------------------------------------------------------
// C[M,64] = A[M,K] @ B[K,64], fp32 WMMA 16x16x4.
// Each wave owns a full 16x64 output strip: one A fragment feeds 4 WMMAs per
// k-step (4x A-traffic reduction vs one-tile-per-wave). 256 threads = 8 waves
// = 128 rows per block.
// A layout (16x4 f32): lanes 0-15 -> K = kb+{0,1}; lanes 16-31 -> K = kb+{2,3}.
// B layout (4x16): VGPR0 row kb+2g, VGPR1 row kb+2g+1 (g = lane>>4).
// C layout: VGPR j -> row j (lanes 0-15) / row j+8 (lanes 16-31).
// ---------------------------------------------------------------------------
__global__ void gat_gemm_wmma(const float* __restrict__ A,
                              const float* __restrict__ B,
                              float* __restrict__ C, int M, int K) {
  const int lane = threadIdx.x & 31;
  const int wave = threadIdx.x >> 5;
  const int g    = lane >> 4;
  const int l16  = lane & 15;
  const int rowBase = blockIdx.x * 128 + wave * 16;
  if (rowBase >= M) return;
  int rowA = rowBase + l16; if (rowA >= M) rowA = M - 1;
  const float* Ap = A + (size_t)rowA * K + 2 * g;
  const float* Bp = B + (size_t)(2 * g) * HID + l16;
  v8f acc0 = {}, acc1 = {}, acc2 = {}, acc3 = {};
  for (int kb = 0; kb < K; kb += 4) {
    v2f a;
    a.x = Ap[kb];
    a.y = Ap[kb + 1];
    const float* br0 = Bp + (size_t)kb * HID;        // B row kb+2g
    const float* br1 = Bp + (size_t)(kb + 1) * HID;  // B row kb+2g+1
    v2f b0, b1, b2, b3;
    b0.x = br0[0];  b0.y = br1[0];
    b1.x = br0[16]; b1.y = br1[16];
    b2.x = br0[32]; b2.y = br1[32];
    b3.x = br0[48]; b3.y = br1[48];
    acc0 = __builtin_amdgcn_wmma_f32_16x16x4_f32(false, a, false, b0, (short)0,
                                                 acc0, false, false);
    acc1 = __builtin_amdgcn_wmma_f32_16x16x4_f32(false, a, false, b1, (short)0,
                                                 acc1, false, false);
    acc2 = __builtin_amdgcn_wmma_f32_16x16x4_f32(false, a, false, b2, (short)0,
                                                 acc2, false, false);
    acc3 = __builtin_amdgcn_wmma_f32_16x16x4_f32(false, a, false, b3, (short)0,
                                                 acc3, false, false);
  }
#pragma unroll
  for (int j = 0; j < 8; ++j) {
    int row = rowBase + 8 * g + j;
    if (row < M) {
      float* Cp = C + (size_t)row * HID + l16;
      Cp[0]  = acc0[j];
      Cp[16] = acc1[j];
      Cp[32] = acc2[j];
      Cp[48] = acc3[j];
    }
  }
}

// a_s[n,h] = <h[n,h,:], att_src[h,:]>, a_d likewise
__global__ void gat_att_proj(const float* __restrict__ h,
                             const float* __restrict__ att_s,
                             const float* __restrict__ att_d,
                             float* __restrict__ as_, float* __restrict__ ad_,
                             int N, int H, int C) {
  long long t = (long long)blockIdx.x * blockDim.x + threadIdx.x;
  if (t >= (long long)N * H) return;
  int n = (int)(t / H), hh = (int)(t % H);
  const float* hp = h + (size_t)n * H * C + (size_t)hh * C;
  float s = 0.f, d = 0.f;
  for (int c = 0; c < C; ++c) {
    float v = hp[c];
    s = fmaf(v, att_s[hh * C + c], s);
    d = fmaf(v, att_d[hh * C + c], d);
  }
  as_[t] = s; ad_[t] = d;
}

// P1: alpha = leaky_relu(a_s[src]+a_d[dst]); store; atomic segment-max on dst
__global__ void gat_edge_alpha(const int* __restrict__ esrc,
                               const int* __restrict__ edst,
                               long long E, int N, int H,
                               const float* __restrict__ as_,
                               const float* __restrict__ ad_,
                               float* __restrict__ alpha,
                               unsigned* __restrict__ amax) {
  long long t = (long long)blockIdx.x * blockDim.x + threadIdx.x;
  long long total = (E + N) * H;
  if (t >= total) return;
  long long e = t / H; int hh = (int)(t % H);
  int s = (e < E) ? esrc[e] : (int)(e - E);
  int d = (e < E) ? edst[e] : (int)(e - E);
  float al = as_[(size_t)s * H + hh] + ad_[(size_t)d * H + hh];
  al = al > 0.f ? al : 0.2f * al;
  alpha[t] = al;
  atomicMax(&amax[(size_t)d * H + hh], f2ord(al));
}

// P2: ex = exp(alpha - max[dst]); store over alpha; atomic-add denom[dst]
__global__ void gat_edge_exp(const int* __restrict__ edst,
                             long long E, int N, int H,
                             float* __restrict__ alpha,
                             const unsigned* __restrict__ amax,
                             float* __restrict__ den) {
  long long t = (long long)blockIdx.x * blockDim.x + threadIdx.x;
  long long total = (E + N) * H;
  if (t >= total) return;
  long long e = t / H; int hh = (int)(t % H);
  int d = (e < E) ? edst[e] : (int)(e - E);
  float m = ord2f(amax[(size_t)d * H + hh]);
  if (!(m == m) || fabsf(m) > 3.0e38f) m = 0.f;  // mirror where(isfinite,...,0)
  float ex = expf(alpha[t] - m);
  alpha[t] = ex;
  atomicAdd(&den[(size_t)d * H + hh], ex);
}

// P3: out[dst,c] += h[src,c] * ex/(den[dst]+1e-16).  64 lanes per edge ->
// coalesced gather of h[src] and coalesced atomics into out[dst]; both the
// 25.6MB feature table and accumulator stay resident in the 192MB L2.
__global__ void gat_edge_msg(const int* __restrict__ esrc,
                             const int* __restrict__ edst,
                             long long E, int N, int H, int cPerHead,
                             const float* __restrict__ hfeat,
                             const float* __restrict__ ex,
                             const float* __restrict__ den,
                             float* __restrict__ out) {
  long long t = (long long)blockIdx.x * blockDim.x + threadIdx.x;
  long long total = (E + N) * HID;
  if (t >= total) return;
  long long e = t >> 6; int c = (int)(t & 63);
  int head = c / cPerHead;
  int s = (e < E) ? esrc[e] : (int)(e - E);
  int d = (e < E) ? edst[e] : (int)(e - E);
  float w = ex[e * H + head] / (den[(size_t)d * H + head] + 1e-16f);
  atomicAdd(&out[(size_t)d * HID + c], hfeat[(size_t)s * HID + c] * w);
}

__global__ void gat_bias_elu(float* __restrict__ out, const float* __restrict__ b,
                             long long total) {
  long long t = (long long)blockIdx.x * blockDim.x + threadIdx.x;
  if (t >= total) return;
  out[t] = elu1(out[t] + b[t & 63]);
}

// h = elu(LN(out2 + b2 + hres)); also pool score s[n] and global max reduce
__global__ void gat_ln_score(float* __restrict__ out2,
                             const float* __restrict__ hres,
                             const float* __restrict__ b2,
                             const float* __restrict__ lng,
                             const float* __restrict__ lnb,
                             const float* __restrict__ wpool,
                             const float* __restrict__ bpool,
                             float* __restrict__ sN,
                             unsigned* __restrict__ smax, int N) {
  int n = blockIdx.x * blockDim.x + threadIdx.x;
  if (n >= N) return;
  float v[HID];
  float mu = 0.f;
#pragma unroll
  for (int c = 0; c < HID; ++c) {
    v[c] = out2[(size_t)n * HID + c] + b2[c] + hres[(size_t)n * HID + c];
    mu += v[c];
  }
  mu *= (1.f / HID);
  float var = 0.f;
#pragma unroll
  for (int c = 0; c < HID; ++c) { float dv = v[c] - mu; var += dv * dv; }
  var *= (1.f / HID);
  float inv = rsqrtf(var + 1e-5f);
  float sacc = bpool[0];
#pragma unroll
  for (int c = 0; c < HID; ++c) {
    float hv = (v[c] - mu) * inv * lng[c] + lnb[c];
    hv = elu1(hv);
    out2[(size_t)n * HID + c] = hv;
    sacc = fmaf(hv, wpool[c], sacc);
  }
  sN[n] = sacc;
  atomicMax(smax, f2ord(sacc));
}

__global__ void gat_pool_exp(float* __restrict__ sN,
                             const unsigned* __restrict__ smax,
                             float* __restrict__ ssum, int N) {
  int n = blockIdx.x * blockDim.x + threadIdx.x;
  if (n >= N) return;
  float ex = expf(sN[n] - ord2f(*smax));
  sN[n] = ex;
  atomicAdd(ssum, ex);
}

__global__ void gat_pool_scatter(const float* __restrict__ h,
                                 const float* __restrict__ sN,
                                 const float* __restrict__ ssum,
                                 const int* __restrict__ batch,
                                 float* __restrict__ pooled, int N) {
  long long t = (long long)blockIdx.x * blockDim.x + threadIdx.x;
  if (t >= (long long)N * HID) return;
  int n = (int)(t >> 6), c = (int)(t & 63);
  float attn = sN[n] / ssum[0];
  atomicAdd(&pooled[(size_t)batch[n] * HID + c], h[(size_t)n * HID + c] * attn);
}

__global__ void gat_fc1(const float* __restrict__ pooled,
                        const float* __restrict__ w, const float* __restrict__ b,
                        float* __restrict__ z, int G) {
  int t = blockIdx.x * blockDim.x + threadIdx.x;
  if (t >= G * HID) return;
  int g = t >> 6, c = t & 63;
  float acc = b[c];
  for (int k = 0; k < HID; ++k) acc = fmaf(pooled[g * HID + k], w[k * HID + c], acc);
  z[t] = elu1(acc);
}

__global__ void gat_fc2(const float* __restrict__ z, const float* __restrict__ w,
                        const float* __restrict__ b, float* __restrict__ out, int G) {
  int g = blockIdx.x * blockDim.x + threadIdx.x;
  if (g >= G) return;
  float acc = b[0];
  for (int k = 0; k < HID; ++k) acc = fmaf(z[g * HID + k], w[k], acc);
  out[g] = acc;
}

static inline unsigned nblk(long long n) { return (unsigned)((n + TPB - 1) / TPB); }

extern "C" void kernel_launch(void* const* d_in, const int* in_sizes, int n_in,
                              void* d_out, int out_size, void* d_ws, size_t ws_size,
                              hipStream_t stream) {
  const float* x      = (const float*)d_in[0];
  const int*   eidx   = (const int*)d_in[1];
  const int*   batch  = (const int*)d_in[2];
  const float* W1     = (const float*)d_in[4];
  const float* att_s1 = (const float*)d_in[5];
  const float* att_d1 = (const float*)d_in[6];
  const float* b1     = (const float*)d_in[7];
  const float* W2     = (const float*)d_in[8];
  const float* att_s2 = (const float*)d_in[9];
  const float* att_d2 = (const float*)d_in[10];
  const float* b2     = (const float*)d_in[11];
  const float* ln_g   = (const float*)d_in[12];
  const float* ln_b   = (const float*)d_in[13];
  const float* w_pool = (const float*)d_in[14];
  const float* b_pool = (const float*)d_in[15];
  const float* w_fc1  = (const float*)d_in[16];
  const float* b_fc1  = (const float*)d_in[17];
  const float* w_fc2  = (const float*)d_in[18];
  const float* b_fc2  = (const float*)d_in[19];

  const long long N  = in_sizes[0] / 128;
  const long long E  = in_sizes[1] / 2;
  const long long Ep = E + N;
  const int G = out_size;
  const int Ni = (int)N;

  const int* esrc = eidx;
  const int* edst = eidx + E;

  float* ws = (float*)d_ws;
  size_t off = 0;
  float* h1   = ws + off; off += (size_t)N * HID;   // conv1 features; reused as h2
  float* out1 = ws + off; off += (size_t)N * HID;   // conv1 out -> h_res
  float* out2 = ws + off; off += (size_t)N * HID;   // conv2 out -> final h
  float* as1  = ws + off; off += (size_t)N * 4;
  float* ad1  = ws + off; off += (size_t)N * 4;
  unsigned* amax1 = (unsigned*)(ws + off); off += (size_t)N * 4;
  float* den1 = ws + off; off += (size_t)N * 4;
  float* as2  = ws + off; off += (size_t)N;
  float* ad2  = ws + off; off += (size_t)N;
  unsigned* amax2 = (unsigned*)(ws + off); off += (size_t)N;
  float* den2 = ws + off; off += (size_t)N;
  float* sN   = ws + off; off += (size_t)N;
  unsigned* redmax = (unsigned*)(ws + off); off += 1;
  float* redsum    = ws + off; off += 1;
  float* pooled = ws + off; off += (size_t)G * HID;
  float* zbuf   = ws + off; off += (size_t)G * HID;
  float* ex1 = ws + off; off += (size_t)Ep * 4;     // alpha then exp, conv1
  float* ex2 = ws + off; off += (size_t)Ep;         // alpha then exp, conv2

  // zero-init accumulators (graph-capture safe)
  hipMemsetAsync(out1, 0, (size_t)N * HID * 4, stream);
  hipMemsetAsync(out2, 0, (size_t)N * HID * 4, stream);
  hipMemsetAsync(amax1, 0, (size_t)N * 4 * 4, stream);
  hipMemsetAsync(den1, 0, (size_t)N * 4 * 4, stream);
  hipMemsetAsync(amax2, 0, (size_t)N * 4, stream);
  hipMemsetAsync(den2, 0, (size_t)N * 4, stream);
  hipMemsetAsync(redmax, 0, 8, stream);             // covers redmax + redsum
  hipMemsetAsync(pooled, 0, (size_t)G * HID * 4, stream);

  const unsigned gemmBlocks = (unsigned)((N + 127) / 128);

  // ---- conv1: GATConv(128 -> 16 x 4 heads) ----
  gat_gemm_wmma<<<gemmBlocks, TPB, 0, stream>>>(x, W1, h1, Ni, 128);
  gat_att_proj<<<nblk(N * 4), TPB, 0, stream>>>(h1, att_s1, att_d1, as1, ad1, Ni, 4, 16);
  gat_edge_alpha<<<nblk(Ep * 4), TPB, 0, stream>>>(esrc, edst, E, Ni, 4, as1, ad1, ex1, amax1);
  gat_edge_exp<<<nblk(Ep * 4), TPB, 0, stream>>>(edst, E, Ni, 4, ex1, amax1, den1);
  gat_edge_msg<<<nblk(Ep * HID), TPB, 0, stream>>>(esrc, edst, E, Ni, 4, 16, h1, ex1, den1, out1);
  gat_bias_elu<<<nblk(N * HID), TPB, 0, stream>>>(out1, b1, N * HID);

  // ---- conv2: GATConv(64 -> 64, 1 head) ----
  gat_gemm_wmma<<<gemmBlocks, TPB, 0, stream>>>(out1, W2, h1, Ni, 64);
  gat_att_proj<<<nblk(N), TPB, 0, stream>>>(h1, att_s2, att_d2, as2, ad2, Ni, 1, 64);
  gat_edge_alpha<<<nblk(Ep), TPB, 0, stream>>>(esrc, edst, E, Ni, 1, as2, ad2, ex2, amax2);
  gat_edge_exp<<<nblk(Ep), TPB, 0, stream>>>(edst, E, Ni, 1, ex2, amax2, den2);
  gat_edge_msg<<<nblk(Ep * HID), TPB, 0, stream>>>(esrc, edst, E, Ni, 1, 64, h1, ex2, den2, out2);

  // ---- residual + LN + elu + pool scores / global softmax / scatter ----
  gat_ln_score<<<nblk(N), TPB, 0, stream>>>(out2, out1, b2, ln_g, ln_b, w_pool,
                                            b_pool, sN, redmax, Ni);
  gat_pool_exp<<<nblk(N), TPB, 0, stream>>>(sN, redmax, redsum, Ni);
  gat_pool_scatter<<<nblk(N * HID), TPB, 0, stream>>>(out2, sN, redsum, batch, pooled, Ni);

  // ---- heads: fc1 (elu) + fc2 ----
  gat_fc1<<<nblk((long long)G * HID), TPB, 0, stream>>>(pooled, w_fc1, b_fc1, zbuf, G);
  gat_fc2<<<nblk(G), TPB, 0, stream>>>(zbuf, w_fc2, b_fc2, (float*)d_out, G);
}